// meta_optimizer_7507602833728
// MI455X (gfx1250) — compile-verified
//
#include <hip/hip_runtime.h>
#include <math.h>

typedef __attribute__((ext_vector_type(2))) float v2f;
typedef __attribute__((ext_vector_type(8))) float v8f;

#define R_DIM 8192
#define C_DIM 2048
#define HID   20
#define S_LR      0.1f
#define P_CLAMP   10.0f
#define OUT_SCALE 0.1f

#define MTILE 64
#define NTILE 128
#define KB    32
#define CB    64   // Cholesky panel width

// ---------------------------------------------------------------------------
// CDNA5 async global->LDS copy (no VGPR staging, tracked by ASYNCcnt)
// ---------------------------------------------------------------------------
__device__ __forceinline__ void async_cp16(void* lds, const void* gmem) {
  asm volatile("global_load_async_to_lds_b128 %0, %1, off"
               :: "v"((__attribute__((address_space(3))) void*)lds),
                  "v"(gmem)
               : "memory");
}

// ---------------------------------------------------------------------------
// NT GEMM: C[m,n] = alpha * sum_k A[m,k]*B[n,k]
// A: MxK (ld=K), B: NxK (ld=K), C: MxN (ld=N).
// 256 threads = 8 waves; 64x128 WG tile; wave = 16x64 (4 f32 WMMA accums);
// double-buffered LDS filled with async-to-LDS loads.
// ---------------------------------------------------------------------------
__global__ __launch_bounds__(256) void gemm_nt_wmma(
    const float* A, const float* B, float* __restrict__ C,
    int M, int N, int K, float alpha)
{
  __shared__ float As[2][MTILE][KB + 4];
  __shared__ float Bs[2][NTILE][KB + 4];
  const int tid  = threadIdx.x;
  const int lane = tid & 31, wv = tid >> 5;
  const int lh   = lane >> 4, l16 = lane & 15;
  const int m0 = blockIdx.y * MTILE, n0 = blockIdx.x * NTILE;
  const int wm = wv & 3, wn = wv >> 2;

  v8f acc[4] = {};

  auto issue = [&](int buf, int k0) {
#pragma unroll
    for (int r = 0; r < 2; ++r) {            // A tile 64x32
      int idx = tid + 256 * r, row = idx >> 3, c4 = (idx & 7) << 2;
      async_cp16(&As[buf][row][c4], A + (size_t)(m0 + row) * K + k0 + c4);
    }
#pragma unroll
    for (int r = 0; r < 4; ++r) {            // B tile 128x32
      int idx = tid + 256 * r, row = idx >> 3, c4 = (idx & 7) << 2;
      async_cp16(&Bs[buf][row][c4], B + (size_t)(n0 + row) * K + k0 + c4);
    }
  };

  const int nk = K / KB;
  issue(0, 0);
  for (int i = 0; i < nk; ++i) {
    const int buf = i & 1;
    if (i + 1 < nk) {
      issue(buf ^ 1, (i + 1) * KB);
      asm volatile("s_wait_asynccnt 6" ::: "memory");   // this buffer's 6 done
    } else {
      asm volatile("s_wait_asynccnt 0" ::: "memory");
    }
    __syncthreads();
#pragma unroll
    for (int kk = 0; kk < KB; kk += 4) {
      const int kb = kk + 2 * lh;            // f32 frag: K = 2*(lane/16) + vgpr
      v2f a;
      a.x = As[buf][16 * wm + l16][kb];
      a.y = As[buf][16 * wm + l16][kb + 1];
#pragma unroll
      for (int t = 0; t < 4; ++t) {
        v2f b;
        b.x = Bs[buf][64 * wn + 16 * t + l16][kb];
        b.y = Bs[buf][64 * wn + 16 * t + l16][kb + 1];
        acc[t] = __builtin_amdgcn_wmma_f32_16x16x4_f32(false, a, false, b,
                                                       (short)0, acc[t], false, false);
      }
    }
    __syncthreads();                         // all waves done reading buf
  }
  const int mr = m0 + 16 * wm + 8 * lh;      // C/D: M = r + 8*(lane/16), N = lane%16
#pragma unroll
  for (int t = 0; t < 4; ++t) {
    const int nc = n0 + 64 * wn + 16 * t + l16;
#pragma unroll
    for (int r = 0; r < 8; ++r)
      C[(size_t)(mr + r) * N + nc] = alpha * acc[t][r];
  }
}

// ---------------------------------------------------------------------------
// NN GEMM with fused epilogue:
// C[m,n] = alpha * sum_k A[m,k]*B[k,n] + beta1*E1[m,n] + beta2*E2[m,n]
// A: MxK (ld=lda), B: KxN (ld=N), C/E: MxN (ld=N). E2 may alias C.
// ---------------------------------------------------------------------------
__global__ __launch_bounds__(256) void gemm_nn_wmma(
    const float* A, int lda, const float* B, float* C,
    const float* E1, float beta1, const float* E2, float beta2,
    int M, int N, int K, float alpha)
{
  __shared__ float As[2][MTILE][KB + 4];     // 64 x 36
  __shared__ float Bs[2][KB][NTILE + 4];     // 32 x 132
  const int tid  = threadIdx.x;
  const int lane = tid & 31, wv = tid >> 5;
  const int lh   = lane >> 4, l16 = lane & 15;
  const int m0 = blockIdx.y * MTILE, n0 = blockIdx.x * NTILE;
  const int wm = wv & 3, wn = wv >> 2;

  v8f acc[4] = {};

  auto issue = [&](int buf, int k0) {
#pragma unroll
    for (int r = 0; r < 2; ++r) {            // A tile 64x32
      int idx = tid + 256 * r, row = idx >> 3, c4 = (idx & 7) << 2;
      async_cp16(&As[buf][row][c4], A + (size_t)(m0 + row) * lda + k0 + c4);
    }
#pragma unroll
    for (int r = 0; r < 4; ++r) {            // B tile 32x128
      int idx = tid + 256 * r, row = idx >> 5, c4 = (idx & 31) << 2;
      async_cp16(&Bs[buf][row][c4], B + (size_t)(k0 + row) * N + n0 + c4);
    }
  };

  const int nk = K / KB;
  issue(0, 0);
  for (int i = 0; i < nk; ++i) {
    const int buf = i & 1;
    if (i + 1 < nk) {
      issue(buf ^ 1, (i + 1) * KB);
      asm volatile("s_wait_asynccnt 6" ::: "memory");
    } else {
      asm volatile("s_wait_asynccnt 0" ::: "memory");
    }
    __syncthreads();
#pragma unroll
    for (int kk = 0; kk < KB; kk += 4) {
      const int kb = kk + 2 * lh;
      v2f a;
      a.x = As[buf][16 * wm + l16][kb];
      a.y = As[buf][16 * wm + l16][kb + 1];
#pragma unroll
      for (int t = 0; t < 4; ++t) {
        v2f b;
        b.x = Bs[buf][kb][64 * wn + 16 * t + l16];
        b.y = Bs[buf][kb + 1][64 * wn + 16 * t + l16];
        acc[t] = __builtin_amdgcn_wmma_f32_16x16x4_f32(false, a, false, b,
                                                       (short)0, acc[t], false, false);
      }
    }
    __syncthreads();
  }
  const int mr = m0 + 16 * wm + 8 * lh;
#pragma unroll
  for (int t = 0; t < 4; ++t) {
    const int nc = n0 + 64 * wn + 16 * t + l16;
#pragma unroll
    for (int r = 0; r < 8; ++r) {
      size_t o = (size_t)(mr + r) * N + nc;
      float v = alpha * acc[t][r];
      if (E1) v += beta1 * E1[o];
      if (E2) v += beta2 * E2[o];
      C[o] = v;
    }
  }
}

// ---------------------------------------------------------------------------
// Small helper kernels
// ---------------------------------------------------------------------------
__global__ void zero2(float* p) { p[0] = 0.0f; p[1] = 0.0f; }

__global__ void symmetrize(float* A, int n) {
  int j = blockIdx.x * 16 + threadIdx.x;
  int i = blockIdx.y * 16 + threadIdx.y;
  if (i < n && j < n && i <= j) {
    float a = A[(size_t)i * n + j], b = A[(size_t)j * n + i];
    float v = 0.5f * (a + b);
    A[(size_t)i * n + j] = v;
    A[(size_t)j * n + i] = v;
  }
}

// out[j] = inv * sum_i W[i][j]^2
__global__ void colsq(const float* __restrict__ W, float* __restrict__ out,
                      int rows, int N, float inv) {
  int j = blockIdx.x * 256 + threadIdx.x;
  float s = 0.f;
  for (int i = 0; i < rows; ++i) { float v = W[(size_t)i * N + j]; s += v * v; }
  out[j] = s * inv;
}

// out[i] = inv * sum_j W[i][j]^2   (one block per row)
__global__ void rowsq(const float* __restrict__ W, float* __restrict__ out,
                      int cols, int N, float inv) {
  __shared__ float red[256];
  int i = blockIdx.x;
  float s = 0.f;
  for (int j = threadIdx.x; j < cols; j += 256) { float v = W[(size_t)i * N + j]; s += v * v; }
  red[threadIdx.x] = s;
  __syncthreads();
  for (int st = 128; st > 0; st >>= 1) {
    if (threadIdx.x < st) red[threadIdx.x] += red[threadIdx.x + st];
    __syncthreads();
  }
  if (threadIdx.x == 0) out[i] = red[0] * inv;
}

__device__ __forceinline__ float sigmoidf(float x) { return 1.0f / (1.0f + expf(-x)); }

// One thread per batch element: x = clamp(log|d|/10,-1,1) -> 2-layer LSTM ->
// head dot -> *0.1; store per-element value; block-reduce sum into accum.
__global__ __launch_bounds__(256) void lstm_head(
    const float* __restrict__ diag, int n,
    const float* __restrict__ h0, const float* __restrict__ c0,   // (2, n, HID)
    const float* __restrict__ Wih0, const float* __restrict__ Whh0,
    const float* __restrict__ bih0, const float* __restrict__ bhh0,
    const float* __restrict__ Wih1, const float* __restrict__ Whh1,
    const float* __restrict__ bih1, const float* __restrict__ bhh1,
    const float* __restrict__ hW, const float* __restrict__ hB,
    float* __restrict__ tmp, float* __restrict__ accum)
{
  __shared__ float red[256];
  int i = blockIdx.x * 256 + threadIdx.x;
  float val = 0.f;
  if (i < n) {
    float d = diag[i];
    float x = logf(fabsf(d)) * (1.0f / P_CLAMP);
    x = fminf(fmaxf(x, -1.0f), 1.0f);
    float hp[HID], cp[HID], h1[HID];
    for (int h = 0; h < HID; ++h) {
      hp[h] = h0[(size_t)i * HID + h];
      cp[h] = c0[(size_t)i * HID + h];
    }
    // layer 0 (input dim 1); gate order i,f,g,o
    for (int g = 0; g < HID; ++g) {
      float gi = Wih0[g]           * x + bih0[g]           + bhh0[g];
      float gf = Wih0[HID + g]     * x + bih0[HID + g]     + bhh0[HID + g];
      float gc = Wih0[2 * HID + g] * x + bih0[2 * HID + g] + bhh0[2 * HID + g];
      float go = Wih0[3 * HID + g] * x + bih0[3 * HID + g] + bhh0[3 * HID + g];
      for (int h = 0; h < HID; ++h) {
        float hv = hp[h];
        gi += Whh0[(size_t)g * HID + h] * hv;
        gf += Whh0[(size_t)(HID + g) * HID + h] * hv;
        gc += Whh0[(size_t)(2 * HID + g) * HID + h] * hv;
        go += Whh0[(size_t)(3 * HID + g) * HID + h] * hv;
      }
      float cn = sigmoidf(gf) * cp[g] + sigmoidf(gi) * tanhf(gc);
      h1[g] = sigmoidf(go) * tanhf(cn);
    }
    // layer 1 (input dim HID)
    const float* h0b = h0 + (size_t)n * HID;
    const float* c0b = c0 + (size_t)n * HID;
    for (int h = 0; h < HID; ++h) {
      hp[h] = h0b[(size_t)i * HID + h];
      cp[h] = c0b[(size_t)i * HID + h];
    }
    float out = hB[0];
    for (int g = 0; g < HID; ++g) {
      float gi = bih1[g] + bhh1[g];
      float gf = bih1[HID + g] + bhh1[HID + g];
      float gc = bih1[2 * HID + g] + bhh1[2 * HID + g];
      float go = bih1[3 * HID + g] + bhh1[3 * HID + g];
      for (int h = 0; h < HID; ++h) {
        gi += Wih1[(size_t)g * HID + h] * h1[h]             + Whh1[(size_t)g * HID + h] * hp[h];
        gf += Wih1[(size_t)(HID + g) * HID + h] * h1[h]     + Whh1[(size_t)(HID + g) * HID + h] * hp[h];
        gc += Wih1[(size_t)(2 * HID + g) * HID + h] * h1[h] + Whh1[(size_t)(2 * HID + g) * HID + h] * hp[h];
        go += Wih1[(size_t)(3 * HID + g) * HID + h] * h1[h] + Whh1[(size_t)(3 * HID + g) * HID + h] * hp[h];
      }
      float cn = sigmoidf(gf) * cp[g] + sigmoidf(gi) * tanhf(gc);
      out += hW[g] * (sigmoidf(go) * tanhf(cn));
    }
    val = out * OUT_SCALE;
    tmp[i] = val;
  }
  red[threadIdx.x] = val;
  __syncthreads();
  for (int st = 128; st > 0; st >>= 1) {
    if (threadIdx.x < st) red[threadIdx.x] += red[threadIdx.x + st];
    __syncthreads();
  }
  if (threadIdx.x == 0) atomicAdd(accum, red[0]);
}

__global__ void finalize_vec(const float* __restrict__ tmp, const float* __restrict__ accum,
                             const float* __restrict__ before, float* __restrict__ outv,
                             int n, float invn) {
  int i = blockIdx.x * 256 + threadIdx.x;
  if (i < n) outv[i] = fmaxf(tmp[i] - accum[0] * invn + 1.0f, before[i]);
}

// PT[i][j] = Rv[i] * GT[i][j] * Lv[j]   (in place, GT is C_DIM x R_DIM)
__global__ void scale_PT(float* __restrict__ W, const float* __restrict__ Rv,
                         const float* __restrict__ Lv) {
  size_t idx = (size_t)blockIdx.x * 256 + threadIdx.x;
  int row = (int)(idx >> 13);        // / R_DIM
  int col = (int)(idx & (R_DIM - 1));
  W[idx] *= Rv[row] * Lv[col];
}

// ---------------------------------------------------------------------------
// Blocked Cholesky (in place, lower triangle of G, n = C_DIM, ld = C_DIM)
// ---------------------------------------------------------------------------
__global__ __launch_bounds__(CB) void potf2(float* G, int k) {
  __shared__ float Ls[CB][CB + 1];
  int t = threadIdx.x;
  const size_t base = (size_t)(k * CB) * C_DIM + k * CB;
  for (int j = 0; j < CB; ++j) Ls[t][j] = G[base + (size_t)t * C_DIM + j];
  __syncthreads();
  for (int j = 0; j < CB; ++j) {
    if (t == j) Ls[j][j] = sqrtf(Ls[j][j]);
    __syncthreads();
    if (t > j) Ls[t][j] /= Ls[j][j];
    __syncthreads();
    if (t > j)
      for (int jj = j + 1; jj <= t; ++jj) Ls[t][jj] -= Ls[t][j] * Ls[jj][j];
    __syncthreads();
  }
  for (int j = 0; j <= t; ++j) G[base + (size_t)t * C_DIM + j] = Ls[t][j];
}

// L21 = A21 * L11^-T : one thread per panel row, forward substitution vs L11
__global__ __launch_bounds__(256) void trsm_panel(float* G, int k, int row0) {
  __shared__ float Ls[CB][CB + 1];
  for (int idx = threadIdx.x; idx < CB * CB; idx += 256) {
    int r = idx / CB, c = idx % CB;
    Ls[r][c] = G[(size_t)(k * CB + r) * C_DIM + k * CB + c];
  }
  __syncthreads();
  int i = row0 + blockIdx.x * 256 + threadIdx.x;
  if (i < C_DIM) {
    float x[CB];
    const size_t base = (size_t)i * C_DIM + k * CB;
    for (int j = 0; j < CB; ++j) {
      float acc = G[base + j];
      for (int t = 0; t < j; ++t) acc -= x[t] * Ls[j][t];
      x[j] = acc / Ls[j][j];
    }
    for (int j = 0; j < CB; ++j) G[base + j] = x[j];
  }
}

// A22(lower) -= L21 * L21^T
__global__ void syrk_update(float* G, int k, int s0) {
  int j = s0 + blockIdx.x * 16 + threadIdx.x;
  int i = s0 + blockIdx.y * 16 + threadIdx.y;
  if (i < C_DIM && j < C_DIM && j <= i) {
    const float* Li = G + (size_t)i * C_DIM + k * CB;
    const float* Lj = G + (size_t)j * C_DIM + k * CB;
    float acc = 0.f;
    for (int t = 0; t < CB; ++t) acc += Li[t] * Lj[t];
    G[(size_t)i * C_DIM + j] -= acc;
  }
}

// X[kb*64 .. +63][:] = L_kk^-1 * rhs  (64-deep forward substitution, 8192 cols)
__global__ __launch_bounds__(256) void trsm_rows(const float* __restrict__ G, int kb,
                                                 const float* __restrict__ rhs,
                                                 float* __restrict__ X) {
  __shared__ float Ls[CB][CB + 1];
  for (int idx = threadIdx.x; idx < CB * CB; idx += 256) {
    int r = idx / CB, c = idx % CB;
    Ls[r][c] = G[(size_t)(kb * CB + r) * C_DIM + kb * CB + c];
  }
  __syncthreads();
  int col = blockIdx.x * 256 + threadIdx.x;
  float x[CB];
  for (int j = 0; j < CB; ++j) {
    float acc = rhs[(size_t)j * R_DIM + col];
    for (int t = 0; t < j; ++t) acc -= Ls[j][t] * x[t];
    x[j] = acc / Ls[j][j];
  }
  for (int j = 0; j < CB; ++j) X[(size_t)(kb * CB + j) * R_DIM + col] = x[j];
}

// ---------------------------------------------------------------------------
extern "C" void kernel_launch(void* const* d_in, const int* in_sizes, int n_in,
                              void* d_out, int out_size, void* d_ws, size_t ws_size,
                              hipStream_t stream) {
  const float* s        = (const float*)d_in[0];   // (2048, 8192) = M^T
  const float* sg       = (const float*)d_in[1];
  const float* L_h0     = (const float*)d_in[2];
  const float* L_c0     = (const float*)d_in[3];
  const float* R_h0     = (const float*)d_in[4];
  const float* R_c0     = (const float*)d_in[5];
  const float* L_before = (const float*)d_in[6];
  const float* R_before = (const float*)d_in[7];
  const float* W_ih0 = (const float*)d_in[8];
  const float* W_hh0 = (const float*)d_in[9];
  const float* b_ih0 = (const float*)d_in[10];
  const float* b_hh0 = (const float*)d_in[11];
  const float* W_ih1 = (const float*)d_in[12];
  const float* W_hh1 = (const float*)d_in[13];
  const float* b_ih1 = (const float*)d_in[14];
  const float* b_hh1 = (const float*)d_in[15];
  const float* L_W = (const float*)d_in[16];
  const float* L_b = (const float*)d_in[17];
  const float* R_W = (const float*)d_in[18];
  const float* R_b = (const float*)d_in[19];

  char* ws = (char*)d_ws;
  float* W0 = (float*)ws;    ws += (size_t)C_DIM * R_DIM * 4;   // 64 MB: GT -> PT -> BT
  float* A0 = (float*)ws;    ws += (size_t)C_DIM * C_DIM * 4;   // 16 MB: Asym / A2sym / Gram->L
  float* T  = (float*)ws;    ws += (size_t)CB * R_DIM * 4;      //  2 MB: solve row-panel temp
  float* diagL = (float*)ws; ws += (size_t)R_DIM * 4;
  float* diagR = (float*)ws; ws += (size_t)C_DIM * 4;
  float* tmpL  = (float*)ws; ws += (size_t)R_DIM * 4;
  float* tmpR  = (float*)ws; ws += (size_t)C_DIM * 4;
  float* Lv    = (float*)ws; ws += (size_t)R_DIM * 4;
  float* Rv    = (float*)ws; ws += (size_t)C_DIM * 4;
  float* accs  = (float*)ws; ws += 2 * 4;
  float* X = (float*)d_out;                                     // new_s (2048 x 8192)

  zero2<<<1, 1, 0, stream>>>(accs);

  // A = s_lr * (s . sg^T)  (2048x2048, K=8192); then symmetrize -> Asym
  gemm_nt_wmma<<<dim3(C_DIM / NTILE, C_DIM / MTILE), 256, 0, stream>>>(
      s, sg, A0, C_DIM, C_DIM, R_DIM, S_LR);
  symmetrize<<<dim3(C_DIM / 16, C_DIM / 16), dim3(16, 16), 0, stream>>>(A0, C_DIM);

  // GT = s_lr*sg - Asym . s   (2048x8192, K=2048)
  gemm_nn_wmma<<<dim3(R_DIM / NTILE, C_DIM / MTILE), 256, 0, stream>>>(
      A0, C_DIM, s, W0, sg, S_LR, nullptr, 0.f, C_DIM, R_DIM, C_DIM, -1.0f);

  // ggt_diag (len r): column sums of GT^2 / c ; gtg_diag (len c): row sums / r
  colsq<<<R_DIM / 256, 256, 0, stream>>>(W0, diagL, C_DIM, R_DIM, 1.0f / (float)C_DIM);
  rowsq<<<C_DIM, 256, 0, stream>>>(W0, diagR, R_DIM, R_DIM, 1.0f / (float)R_DIM);

  // per-element 2-layer LSTMs + linear head, with mean reduction
  lstm_head<<<R_DIM / 256, 256, 0, stream>>>(diagL, R_DIM, L_h0, L_c0,
      W_ih0, W_hh0, b_ih0, b_hh0, W_ih1, W_hh1, b_ih1, b_hh1, L_W, L_b, tmpL, accs + 0);
  lstm_head<<<C_DIM / 256, 256, 0, stream>>>(diagR, C_DIM, R_h0, R_c0,
      W_ih0, W_hh0, b_ih0, b_hh0, W_ih1, W_hh1, b_ih1, b_hh1, R_W, R_b, tmpR, accs + 1);
  finalize_vec<<<R_DIM / 256, 256, 0, stream>>>(tmpL, accs + 0, L_before, Lv, R_DIM, 1.0f / R_DIM);
  finalize_vec<<<C_DIM / 256, 256, 0, stream>>>(tmpR, accs + 1, R_before, Rv, C_DIM, 1.0f / C_DIM);

  // PT = diag(Rv) . GT . diag(Lv)  in place
  scale_PT<<<(unsigned)((size_t)C_DIM * R_DIM / 256), 256, 0, stream>>>(W0, Rv, Lv);

  // A2 = s . PT^T (2048x2048, K=8192); symmetrize
  gemm_nt_wmma<<<dim3(C_DIM / NTILE, C_DIM / MTILE), 256, 0, stream>>>(
      s, W0, A0, C_DIM, C_DIM, R_DIM, 1.0f);
  symmetrize<<<dim3(C_DIM / 16, C_DIM / 16), dim3(16, 16), 0, stream>>>(A0, C_DIM);

  // BT = A2sym . s + s - PT   (in place over PT; B = M - Pnew)
  gemm_nn_wmma<<<dim3(R_DIM / NTILE, C_DIM / MTILE), 256, 0, stream>>>(
      A0, C_DIM, s, W0, s, 1.0f, W0, -1.0f, C_DIM, R_DIM, C_DIM, 1.0f);

  // Gram = BT . BT^T (2048x2048, K=8192)
  gemm_nt_wmma<<<dim3(C_DIM / NTILE, C_DIM / MTILE), 256, 0, stream>>>(
      W0, W0, A0, C_DIM, C_DIM, R_DIM, 1.0f);

  // Cholesky (CholeskyQR == reduced QR with positive-diag sign convention)
  for (int k = 0; k < C_DIM / CB; ++k) {
    potf2<<<1, CB, 0, stream>>>(A0, k);
    int row0 = (k + 1) * CB, nr = C_DIM - row0;
    if (nr > 0) {
      trsm_panel<<<(nr + 255) / 256, 256, 0, stream>>>(A0, k, row0);
      syrk_update<<<dim3((nr + 15) / 16, (nr + 15) / 16), dim3(16, 16), 0, stream>>>(A0, k, row0);
    }
  }

  // new_s = Q^T = L^-1 . BT, blocked: WMMA update + 64-deep substitution
  for (int kb = 0; kb < C_DIM / CB; ++kb) {
    const float* rhs;
    if (kb == 0) {
      rhs = W0;                                   // BT rows 0..63
    } else {
      // T = BT_kb - L[kb, 0:kb] . X[0:kb, :]   (M=64, N=8192, K=kb*64)
      gemm_nn_wmma<<<dim3(R_DIM / NTILE, 1), 256, 0, stream>>>(
          A0 + (size_t)kb * CB * C_DIM, C_DIM, X, T,
          W0 + (size_t)kb * CB * R_DIM, 1.0f, nullptr, 0.f,
          CB, R_DIM, kb * CB, -1.0f);
      rhs = T;
    }
    trsm_rows<<<R_DIM / 256, 256, 0, stream>>>(A0, kb, rhs, X);
  }
}